// WritingLayer_13563506721323
// MI455X (gfx1250) — compile-verified
//
#include <hip/hip_runtime.h>
#include <hip/hip_bf16.h>

typedef __attribute__((ext_vector_type(2))) float v2f;
typedef __attribute__((ext_vector_type(8))) float v8f;

#define HID   256
#define DIN   128
#define NB    64
#define NS    512
#define CT    16      // timesteps per chunk in the val scan

// decay constants: exp(-1/20) for both mem and trace
__device__ constexpr float DM = 0.95122942450071400910f;
__device__ constexpr float OM = 1.0f - 0.95122942450071400910f;
__device__ constexpr float DT = 0.95122942450071400910f;
__device__ constexpr float OT = 1.0f - 0.95122942450071400910f;

// d_out flat layout (floats): mem, key_seq, val_seq, key_state, val_state, key_trace, val_trace
#define MEM_OFF        ((size_t)0)
#define KEY_SEQ_OFF    ((size_t)4194304)
#define VAL_SEQ_OFF    ((size_t)12582912)
#define KEY_STATE_OFF  ((size_t)20971520)
#define VAL_STATE_OFF  ((size_t)20987904)
#define KEY_TRACE_OFF  ((size_t)21004288)
#define VAL_TRACE_OFF  ((size_t)21020672)

// ---------------------------------------------------------------------------
// Kernel A: i = x @ W^T  (M=32768, N=512, K=128) using f32 WMMA 16x16x4.
// Columns [0,256) -> ik (workspace); columns [256,512) -> iv staged in d_out
// val_seq slot (overwritten in place by the val scan later).
// ---------------------------------------------------------------------------
__global__ __launch_bounds__(256)
void gemm_in_kernel(const float* __restrict__ X, const float* __restrict__ W,
                    float* __restrict__ ik_ws, float* __restrict__ out) {
  const int lane = threadIdx.x & 31;
  const int wave = threadIdx.x >> 5;
  const int tile = blockIdx.x * 8 + wave;   // 65536 tiles total
  const int tn = tile & 31;                 // 32 col tiles (N=512)
  const int tm = tile >> 5;                 // 2048 row tiles (M=32768)
  const int lo = lane & 15;
  const int hi = lane >> 4;
  const int m0 = tm * 16;
  const int n0 = tn * 16;

  const float* xrow = X + (size_t)(m0 + lo) * DIN;
  const float* wrow = W + (size_t)(n0 + lo) * DIN;

  v8f c = {0.f, 0.f, 0.f, 0.f, 0.f, 0.f, 0.f, 0.f};
#pragma unroll
  for (int kk = 0; kk < DIN; kk += 4) {
    v2f a = *(const v2f*)(xrow + kk + 2 * hi);
    v2f b = *(const v2f*)(wrow + kk + 2 * hi);
    c = __builtin_amdgcn_wmma_f32_16x16x4_f32(false, a, false, b,
                                              (short)0, c, false, false);
  }

  const long n = n0 + lo;
#pragma unroll
  for (int r = 0; r < 8; ++r) {
    const size_t m = (size_t)(m0 + r + 8 * hi);
    const float v = c[r];
    if (n < HID) {
      ik_ws[m * HID + n] = v;
    } else {
      out[VAL_SEQ_OFF + m * HID + (n - HID)] = v;
    }
  }
}

// ---------------------------------------------------------------------------
// Kernel B: key-path scan (parallel over b,h; sequential over t).
// ---------------------------------------------------------------------------
__global__ __launch_bounds__(256)
void key_scan_kernel(float* __restrict__ ik_kt, float* __restrict__ out) {
  const int b = blockIdx.x;
  const int h = threadIdx.x;
  float ks = 0.f, ktr = 0.f;
  const size_t base = (size_t)b * NS * HID + h;
  for (int t = 0; t < NS; ++t) {
    const size_t idx = base + (size_t)t * HID;
    const float ik = ik_kt[idx];
    ks = DM * ks + OM * ik;
    const float ko = tanhf(ks);
    out[KEY_SEQ_OFF + idx] = ko;
    ktr = DT * ktr + OT * ko;
    ik_kt[idx] = ktr;                  // kt overwrites ik in-place
  }
  out[KEY_STATE_OFF + (size_t)b * HID + h] = ks;
  out[KEY_TRACE_OFF + (size_t)b * HID + h] = ktr;
}

// ---------------------------------------------------------------------------
// Kernel C: chunked WMMA val scan.
// One WG = batch b, 32-row block of mem (8 blocks/batch -> 512 WGs),
// 64 threads = 2 waves, one 16-row WMMA M-tile per wave.
// Per 16-step chunk:
//   P  = mem @ Ko_chunk^T            (WMMA, K=256)
//   G  = Ko_chunk @ Kt_chunk^T       (WMMA, K=256, wave 0)
//   sequential 16 steps: ikv_t = 0.2*P[:,t] + 0.1*sum_{tau<t} G[t,tau]*vt_tau
//   mem += 0.5 * Vt^T @ Kt_chunk     (WMMA, K=16, C-fragment = mem tile)
// ---------------------------------------------------------------------------
#define RB    32       // rows of mem per WG
#define MSTR  258      // LDS row stride for mem (even -> aligned v2f reads)
#define KSTR  260      // LDS row stride for ko/kt chunk [t][h]

__global__ __launch_bounds__(64)
void val_scan_kernel(const float* __restrict__ kt_ws, float* __restrict__ out) {
  __shared__ float s_mem[RB * MSTR];     // fast-weight rows
  __shared__ float s_koc[CT * KSTR];     // ko chunk  [t][h]
  __shared__ float s_ktc[CT * KSTR];     // kt chunk  [t][h]
  __shared__ float s_P [CT * 33];        // P   [t][row]
  __shared__ float s_G [CT * 17];        // G   [t][tau]
  __shared__ float s_vt[CT * 33];        // vt  [t][row]
  __shared__ float s_iv[CT * 32];        // iv  [t][row]
  __shared__ float s_vo[CT * 32];        // vo  [t][row]

  const int b   = blockIdx.x >> 3;
  const int blk = blockIdx.x & 7;
  const int tid = threadIdx.x;
  const int lane = tid & 31;
  const int wid  = tid >> 5;            // wave id: M-tile
  const int lo = lane & 15;
  const int hi = lane >> 4;
  const int gr0 = blk * RB;             // first global hidden row of this WG
  const size_t seqbase = (size_t)b * NS * HID;

  for (int i = tid; i < RB * MSTR; i += 64) s_mem[i] = 0.f;
  float vs = 0.f, vt = 0.f;
  __syncthreads();

  for (int c0 = 0; c0 < NS; c0 += CT) {
    const size_t cb = seqbase + (size_t)c0 * HID;

    // ---- stage ko/kt/iv chunk (coalesced) ----
    for (int idx = tid; idx < CT * HID; idx += 64) {
      const int t = idx >> 8, h = idx & 255;
      const size_t g = cb + (size_t)t * HID + h;
      s_koc[t * KSTR + h] = out[KEY_SEQ_OFF + g];
      s_ktc[t * KSTR + h] = kt_ws[g];
    }
    for (int idx = tid; idx < CT * RB; idx += 64) {
      const int t = idx >> 5, i = idx & 31;
      s_iv[idx] = out[VAL_SEQ_OFF + cb + (size_t)t * HID + gr0 + i];
    }
    __syncthreads();

    // ---- G = Ko @ Kt^T (wave 0 only; 16x16, K=256) ----
    if (wid == 0) {
      v8f g = {0.f, 0.f, 0.f, 0.f, 0.f, 0.f, 0.f, 0.f};
#pragma unroll 16
      for (int kk = 0; kk < HID; kk += 4) {
        v2f a, bb;
        a.x  = s_koc[lo * KSTR + kk + 2 * hi];
        a.y  = s_koc[lo * KSTR + kk + 2 * hi + 1];
        bb.x = s_ktc[lo * KSTR + kk + 2 * hi];
        bb.y = s_ktc[lo * KSTR + kk + 2 * hi + 1];
        g = __builtin_amdgcn_wmma_f32_16x16x4_f32(false, a, false, bb,
                                                  (short)0, g, false, false);
      }
#pragma unroll
      for (int r = 0; r < 8; ++r) s_G[(r + 8 * hi) * 17 + lo] = g[r];
    }

    // ---- P = mem @ Ko^T (each wave: its 16-row tile; 16x16, K=256) ----
    {
      const int m = wid * 16 + lo;
      v8f p = {0.f, 0.f, 0.f, 0.f, 0.f, 0.f, 0.f, 0.f};
#pragma unroll 16
      for (int kk = 0; kk < HID; kk += 4) {
        v2f a, bb;
        a.x  = s_mem[m * MSTR + kk + 2 * hi];
        a.y  = s_mem[m * MSTR + kk + 2 * hi + 1];
        bb.x = s_koc[lo * KSTR + kk + 2 * hi];
        bb.y = s_koc[lo * KSTR + kk + 2 * hi + 1];
        p = __builtin_amdgcn_wmma_f32_16x16x4_f32(false, a, false, bb,
                                                  (short)0, p, false, false);
      }
#pragma unroll
      for (int r = 0; r < 8; ++r)
        s_P[lo * 33 + wid * 16 + r + 8 * hi] = p[r];
    }
    __syncthreads();

    // ---- sequential 16 steps (wave 0; one lane per row) ----
    if (tid < 32) {
      for (int t = 0; t < CT; ++t) {
        float corr = 0.f;
        for (int tau = 0; tau < t; ++tau)
          corr += s_G[t * 17 + tau] * s_vt[tau * 33 + tid];
        const float ikv = 0.2f * s_P[t * 33 + tid] + 0.1f * corr;
        vs = DM * vs + OM * (s_iv[t * 32 + tid] + ikv);
        const float vo = tanhf(vs);
        s_vo[t * 32 + tid] = vo;
        vt = DT * vt + OT * vo;
        s_vt[t * 33 + tid] = vt;
      }
    }
    __syncthreads();

    // ---- write vo chunk to val_seq (coalesced) ----
    for (int idx = tid; idx < CT * RB; idx += 64) {
      const int t = idx >> 5, i = idx & 31;
      out[VAL_SEQ_OFF + cb + (size_t)t * HID + gr0 + i] = s_vo[idx];
    }

    // ---- mem += 0.5 * Vt^T @ Kt (each wave: 16 rows x 256 cols, K=16) ----
#pragma unroll 4
    for (int nt = 0; nt < 16; ++nt) {
      const int n0 = nt * 16;
      v8f cf;
#pragma unroll
      for (int r = 0; r < 8; ++r)
        cf[r] = s_mem[(wid * 16 + r + 8 * hi) * MSTR + n0 + lo];
#pragma unroll
      for (int kk = 0; kk < CT; kk += 4) {
        v2f a, bb;
        a.x  = 0.5f * s_vt[(kk + 2 * hi)     * 33 + wid * 16 + lo];
        a.y  = 0.5f * s_vt[(kk + 2 * hi + 1) * 33 + wid * 16 + lo];
        bb.x = s_ktc[(kk + 2 * hi)     * KSTR + n0 + lo];
        bb.y = s_ktc[(kk + 2 * hi + 1) * KSTR + n0 + lo];
        cf = __builtin_amdgcn_wmma_f32_16x16x4_f32(false, a, false, bb,
                                                   (short)0, cf, false, false);
      }
#pragma unroll
      for (int r = 0; r < 8; ++r)
        s_mem[(wid * 16 + r + 8 * hi) * MSTR + n0 + lo] = cf[r];
    }
    __syncthreads();   // update visible before next chunk's P / re-staging
  }

  // ---- write back this row block of mem ----
  for (int idx = tid; idx < RB * HID; idx += 64) {
    const int r = idx >> 8, j = idx & 255;
    out[MEM_OFF + ((size_t)b * HID + gr0 + r) * HID + j] = s_mem[r * MSTR + j];
  }
  if (tid < 32) {
    out[VAL_STATE_OFF + (size_t)b * HID + gr0 + tid] = vs;
    out[VAL_TRACE_OFF + (size_t)b * HID + gr0 + tid] = vt;
  }
}

// ---------------------------------------------------------------------------
extern "C" void kernel_launch(void* const* d_in, const int* in_sizes, int n_in,
                              void* d_out, int out_size, void* d_ws, size_t ws_size,
                              hipStream_t stream) {
  const float* x = (const float*)d_in[0];   // (64, 512, 128) f32
  const float* W = (const float*)d_in[1];   // (512, 128) f32
  float* out = (float*)d_out;
  float* ws  = (float*)d_ws;                // needs 8,388,608 floats (33.6 MB)

  gemm_in_kernel<<<dim3(8192), dim3(256), 0, stream>>>(x, W, ws, out);
  key_scan_kernel<<<dim3(NB), dim3(256), 0, stream>>>(ws, out);
  val_scan_kernel<<<dim3(NB * 8), dim3(64), 0, stream>>>(ws, out);
}